// GCN_38603166057036
// MI455X (gfx1250) — compile-verified
//
#include <hip/hip_runtime.h>
#include <hip/hip_bf16.h>

// ---------------------------------------------------------------------------
// GCN pipeline for MI455X (gfx1250, wave32).
// Dense GEMMs use V_WMMA_F32_16X16X4_F32 (exact fp32 matrix pipe).
// Edge scatter uses native f32 global atomics (L2-resident, 192MB L2).
// ---------------------------------------------------------------------------

typedef __attribute__((ext_vector_type(2))) float v2f;
typedef __attribute__((ext_vector_type(8))) float v8f;

#define NN    50000
#define CIN   10
#define CH    128
#define CO    256
#define EPSLN 1e-5f
#define EPSGN 1e-5f
#define EPSL2 1e-12f

// ---------------- stage 1: h = relu(LN(x @ W_in + b_in)) -------------------
// one wave per row; lane handles channels lane, lane+32, lane+64, lane+96
__global__ __launch_bounds__(256) void k_input(const float* __restrict__ x,
                                               const float* __restrict__ W,
                                               const float* __restrict__ bias,
                                               const float* __restrict__ g,
                                               const float* __restrict__ bln,
                                               float* __restrict__ h) {
  const int lane = threadIdx.x & 31;
  const int row  = blockIdx.x * 8 + (threadIdx.x >> 5);
  if (row >= NN) return;
  float xr[CIN];
#pragma unroll
  for (int k = 0; k < CIN; ++k) xr[k] = x[(size_t)row * CIN + k];
  float v[4];
#pragma unroll
  for (int j = 0; j < 4; ++j) {
    const int c = lane + 32 * j;
    float s = bias[c];
#pragma unroll
    for (int k = 0; k < CIN; ++k) s = fmaf(xr[k], W[k * CH + c], s);
    v[j] = s;
  }
  float sum = v[0] + v[1] + v[2] + v[3];
  float sq  = v[0]*v[0] + v[1]*v[1] + v[2]*v[2] + v[3]*v[3];
#pragma unroll
  for (int m = 16; m >= 1; m >>= 1) {
    sum += __shfl_xor(sum, m, 32);
    sq  += __shfl_xor(sq,  m, 32);
  }
  const float mu   = sum * (1.0f / CH);
  const float var  = sq * (1.0f / CH) - mu * mu;
  const float rstd = rsqrtf(var + EPSLN);
#pragma unroll
  for (int j = 0; j < 4; ++j) {
    const int c = lane + 32 * j;
    const float t = (v[j] - mu) * rstd * g[c] + bln[c];
    h[(size_t)row * CH + c] = fmaxf(t, 0.0f);
  }
}

// ---------------- fp32 WMMA GEMM: C[M,N] = A[M,K] @ B[K,N] -----------------
// wave computes a 16-row x 64-col tile; K multiple of 4; M multiple of 16.
__global__ __launch_bounds__(128) void k_wmma_gemm(const float* __restrict__ A,
                                                   const float* __restrict__ B,
                                                   float* __restrict__ C,
                                                   int M, int K, int N) {
  const int lane   = threadIdx.x & 31;
  const int waveId = blockIdx.x * 4 + (threadIdx.x >> 5);
  const int colGroups = N >> 6;                  // 64 cols per wave
  const int rowTile   = waveId / colGroups;
  const int cg        = waveId - rowTile * colGroups;
  const int m0 = rowTile * 16;
  if (m0 >= M) return;
  const int n0  = cg * 64;
  const int hi  = lane >> 4;                     // 0: K-pair {0,1}; 1: {2,3}
  const int l15 = lane & 15;

  v8f acc0 = {}, acc1 = {}, acc2 = {}, acc3 = {};
  const float* __restrict__ arow = A + (size_t)(m0 + l15) * K + 2 * hi;

  for (int k = 0; k < K; k += 4) {
    v2f a;
    a.x = arow[k];
    a.y = arow[k + 1];
    const float* __restrict__ bp0 = B + (size_t)(k + 2 * hi)     * N + n0 + l15;
    const float* __restrict__ bp1 = B + (size_t)(k + 2 * hi + 1) * N + n0 + l15;
    v2f b0, b1, b2, b3;
    b0.x = bp0[0];  b0.y = bp1[0];
    b1.x = bp0[16]; b1.y = bp1[16];
    b2.x = bp0[32]; b2.y = bp1[32];
    b3.x = bp0[48]; b3.y = bp1[48];
    acc0 = __builtin_amdgcn_wmma_f32_16x16x4_f32(false, a, false, b0, (short)0, acc0, false, false);
    acc1 = __builtin_amdgcn_wmma_f32_16x16x4_f32(false, a, false, b1, (short)0, acc1, false, false);
    acc2 = __builtin_amdgcn_wmma_f32_16x16x4_f32(false, a, false, b2, (short)0, acc2, false, false);
    acc3 = __builtin_amdgcn_wmma_f32_16x16x4_f32(false, a, false, b3, (short)0, acc3, false, false);
  }
#pragma unroll
  for (int r = 0; r < 8; ++r) {
    const int row = m0 + r + 8 * hi;
    float* __restrict__ crow = C + (size_t)row * N + n0 + l15;
    crow[0]  = acc0[r];
    crow[16] = acc1[r];
    crow[32] = acc2[r];
    crow[48] = acc3[r];
  }
}

// ---------------- degree / normalization -----------------------------------
__global__ void k_deg_init(float* __restrict__ deg) {
  const int i = blockIdx.x * blockDim.x + threadIdx.x;
  if (i < NN) deg[i] = 1.0f;                      // self-loop
}
__global__ void k_deg_edges(const int* __restrict__ dst, float* __restrict__ deg, int E) {
  const int e = blockIdx.x * blockDim.x + threadIdx.x;
  if (e < E) unsafeAtomicAdd(&deg[dst[e]], 1.0f);
}
__global__ void k_dinv(float* __restrict__ deg) {
  const int i = blockIdx.x * blockDim.x + threadIdx.x;
  if (i < NN) deg[i] = rsqrtf(deg[i]);            // deg >= 1 always
}

// ---------------- agg init: self-loop term + bias --------------------------
__global__ void k_agg_init(const float* __restrict__ hw, const float* __restrict__ dinv,
                           const float* __restrict__ bc, float* __restrict__ agg) {
  const size_t idx = (size_t)blockIdx.x * blockDim.x + threadIdx.x;
  if (idx >= (size_t)NN * CH) return;
  const int i = (int)(idx >> 7);
  const int c = (int)(idx & 127);
  const float di = dinv[i];
  agg[idx] = hw[idx] * di * di + bc[c];
}

// ---------------- edge scatter: agg[dst] += hw[src] * dinv[src]*dinv[dst] --
__global__ __launch_bounds__(256) void k_scatter(const int* __restrict__ src,
                                                 const int* __restrict__ dst,
                                                 const float* __restrict__ hw,
                                                 const float* __restrict__ dinv,
                                                 float* __restrict__ agg, int E) {
  const int lane = threadIdx.x & 31;
  const int e    = blockIdx.x * 8 + (threadIdx.x >> 5);
  if (e >= E) return;
  const int s = src[e];
  const int d = dst[e];
  const float nrm = dinv[s] * dinv[d];
  const float* __restrict__ hs = hw + (size_t)s * CH;
  float* __restrict__ ad = agg + (size_t)d * CH;
#pragma unroll
  for (int j = 0; j < 4; ++j) {
    const int c = lane + 32 * j;
    unsafeAtomicAdd(&ad[c], hs[c] * nrm);
  }
}

// ---------------- GraphNorm stats ------------------------------------------
__global__ void k_zero_sums(float* __restrict__ sums) {
  const int i = blockIdx.x * blockDim.x + threadIdx.x;
  if (i < 2 * CH) sums[i] = 0.0f;
}
__global__ __launch_bounds__(128) void k_gn_stats(const float* __restrict__ agg,
                                                  float* __restrict__ sums) {
  const int c  = threadIdx.x;                     // channel
  const int nb = gridDim.x;
  const int rowsPer = (NN + nb - 1) / nb;
  const int r0 = blockIdx.x * rowsPer;
  const int r1 = min(r0 + rowsPer, NN);
  float s = 0.0f, q = 0.0f;
  for (int r = r0; r < r1; ++r) {
    const float v = agg[(size_t)r * CH + c];
    s += v;
    q += v * v;
  }
  unsafeAtomicAdd(&sums[c], s);
  unsafeAtomicAdd(&sums[CH + c], q);
}

// ---------------- GraphNorm apply + ReLU + residual ------------------------
__global__ void k_gn_apply(const float* __restrict__ agg, const float* __restrict__ h,
                           const float* __restrict__ sums, const float* __restrict__ alpha,
                           const float* __restrict__ gw, const float* __restrict__ gb,
                           float* __restrict__ h2) {
  const size_t idx = (size_t)blockIdx.x * blockDim.x + threadIdx.x;
  if (idx >= (size_t)NN * CH) return;
  const int c = (int)(idx & 127);
  const float invN = 1.0f / (float)NN;
  const float mu  = sums[c] * invN;
  const float ex2 = sums[CH + c] * invN;
  const float al  = alpha[c];
  const float var = ex2 - 2.0f * al * mu * mu + al * al * mu * mu;
  const float rstd = rsqrtf(var + EPSGN);
  const float val = (agg[idx] - al * mu) * rstd * gw[c] + gb[c];
  h2[idx] = fmaxf(val, 0.0f) + h[idx];
}

// ---------------- final LN(256) + L2 normalize -----------------------------
// one wave per row; 8 channels per lane
__global__ __launch_bounds__(256) void k_ln2_l2(const float* __restrict__ zpre,
                                                const float* __restrict__ bp,
                                                const float* __restrict__ g,
                                                const float* __restrict__ bln,
                                                float* __restrict__ out) {
  const int lane = threadIdx.x & 31;
  const int row  = blockIdx.x * 8 + (threadIdx.x >> 5);
  if (row >= NN) return;
  float v[8];
  float sum = 0.0f, sq = 0.0f;
#pragma unroll
  for (int j = 0; j < 8; ++j) {
    const int c = lane + 32 * j;
    v[j] = zpre[(size_t)row * CO + c] + bp[c];
    sum += v[j];
    sq  += v[j] * v[j];
  }
#pragma unroll
  for (int m = 16; m >= 1; m >>= 1) {
    sum += __shfl_xor(sum, m, 32);
    sq  += __shfl_xor(sq,  m, 32);
  }
  const float mu   = sum * (1.0f / CO);
  const float var  = sq * (1.0f / CO) - mu * mu;
  const float rstd = rsqrtf(var + EPSLN);
  float n2 = 0.0f;
#pragma unroll
  for (int j = 0; j < 8; ++j) {
    const int c = lane + 32 * j;
    v[j] = (v[j] - mu) * rstd * g[c] + bln[c];
    n2 += v[j] * v[j];
  }
#pragma unroll
  for (int m = 16; m >= 1; m >>= 1) n2 += __shfl_xor(n2, m, 32);
  const float scale = 1.0f / fmaxf(sqrtf(n2), EPSL2);
#pragma unroll
  for (int j = 0; j < 8; ++j) {
    const int c = lane + 32 * j;
    out[(size_t)row * CO + c] = v[j] * scale;
  }
}

// ---------------------------------------------------------------------------
extern "C" void kernel_launch(void* const* d_in, const int* in_sizes, int n_in,
                              void* d_out, int out_size, void* d_ws, size_t ws_size,
                              hipStream_t stream) {
  const float* x     = (const float*)d_in[0];
  const int*   ei    = (const int*)  d_in[1];
  const float* W_in  = (const float*)d_in[2];
  const float* b_in  = (const float*)d_in[3];
  const float* ln1_g = (const float*)d_in[4];
  const float* ln1_b = (const float*)d_in[5];
  const float* Wc    = (const float*)d_in[6];
  const float* bc    = (const float*)d_in[7];
  const float* gn_w  = (const float*)d_in[8];
  const float* gn_b  = (const float*)d_in[9];
  const float* gn_a  = (const float*)d_in[10];
  const float* Wp    = (const float*)d_in[11];
  const float* bp    = (const float*)d_in[12];
  const float* ln2_g = (const float*)d_in[13];
  const float* ln2_b = (const float*)d_in[14];
  float* out = (float*)d_out;

  const int E = in_sizes[1] / 2;
  const int* src = ei;
  const int* dst = ei + E;

  // workspace layout (bytes)
  char* base = (char*)d_ws;
  const size_t SZ_H = (size_t)NN * CH * sizeof(float);   // 25.6 MB
  float* h    = (float*)(base);                          // [0, 25.6M)
  float* agg  = (float*)(base + SZ_H);                   // [25.6M, 51.2M)
  float* hw   = (float*)(base + 2 * SZ_H);               // [51.2M, 76.8M)  also h2
  float* deg  = (float*)(base + 3 * SZ_H);               // 200 KB (dinv in place)
  float* sums = (float*)(base + 3 * SZ_H + (204800));    // 1 KB
  float* zpre = (float*)(base);                          // reuses h+agg (51.2 MB)
  float* h2   = hw;                                      // reuses hw after scatter

  // 1) input transform: x -> h
  k_input<<<(NN + 7) / 8, 256, 0, stream>>>(x, W_in, b_in, ln1_g, ln1_b, h);

  // 2) degrees -> dinv
  k_deg_init<<<(NN + 255) / 256, 256, 0, stream>>>(deg);
  k_deg_edges<<<(E + 255) / 256, 256, 0, stream>>>(dst, deg, E);
  k_dinv<<<(NN + 255) / 256, 256, 0, stream>>>(deg);

  // 3) hw = h @ Wc   (WMMA f32, M=50000, K=128, N=128)
  {
    const int waves = (NN / 16) * (CH / 64);
    k_wmma_gemm<<<(waves + 3) / 4, 128, 0, stream>>>(h, Wc, hw, NN, CH, CH);
  }

  // 4) agg = self-loop + bias, then edge scatter-add
  {
    const size_t tot = (size_t)NN * CH;
    k_agg_init<<<(int)((tot + 255) / 256), 256, 0, stream>>>(hw, deg, bc, agg);
    k_scatter<<<(E + 7) / 8, 256, 0, stream>>>(src, dst, hw, deg, agg, E);
  }

  // 5) GraphNorm stats + apply (+ReLU + residual) -> h2
  k_zero_sums<<<1, 256, 0, stream>>>(sums);
  k_gn_stats<<<200, 128, 0, stream>>>(agg, sums);
  {
    const size_t tot = (size_t)NN * CH;
    k_gn_apply<<<(int)((tot + 255) / 256), 256, 0, stream>>>(agg, h, sums, gn_a,
                                                             gn_w, gn_b, h2);
  }

  // 6) zpre = h2 @ Wp  (WMMA f32, M=50000, K=128, N=256)
  {
    const int waves = (NN / 16) * (CO / 64);
    k_wmma_gemm<<<(waves + 3) / 4, 128, 0, stream>>>(h2, Wp, zpre, NN, CH, CO);
  }

  // 7) LN(256) + L2 normalize -> out
  k_ln2_l2<<<(NN + 7) / 8, 256, 0, stream>>>(zpre, bp, ln2_g, ln2_b, out);

  (void)n_in; (void)out_size; (void)ws_size;
}